// Tree_Net_8624294331015
// MI455X (gfx1250) — compile-verified
//
#include <hip/hip_runtime.h>
#include <math.h>

// CDNA5 / gfx1250: wave32, WMMA f32 16x16x4 (full-precision matrix path).
typedef __attribute__((ext_vector_type(2))) float v2f;
typedef __attribute__((ext_vector_type(8))) float v8f;

#define DIM 1024  // embedding dimension D (reference: 1024)

__device__ __forceinline__ float waveReduceSum(float v) {
  #pragma unroll
  for (int o = 16; o > 0; o >>= 1) v += __shfl_down(v, o, 32);
  return v;
}

// ---------------------------------------------------------------------------
// Kernel 1: gather leaf embeddings and L2-normalize. One block per leaf.
// ---------------------------------------------------------------------------
__global__ void leaf_kernel(const int* __restrict__ leaf_info,
                            const float* __restrict__ emb_w,
                            float* __restrict__ nodes) {
  __shared__ float red[8];
  __shared__ float rinv_s;
  const int leaf = blockIdx.x;
  const int dst = leaf_info[2 * leaf + 0];
  const int vid = leaf_info[2 * leaf + 1];
  const float* src = emb_w + (size_t)vid * DIM;

  float ss = 0.0f;
  for (int d = threadIdx.x; d < DIM; d += blockDim.x) {
    float v = src[d];
    ss += v * v;
  }
  ss = waveReduceSum(ss);
  if ((threadIdx.x & 31) == 0) red[threadIdx.x >> 5] = ss;
  __syncthreads();
  if (threadIdx.x == 0) {
    float t = 0.0f;
    #pragma unroll
    for (int i = 0; i < 8; ++i) t += red[i];
    rinv_s = 1.0f / sqrtf(t);
  }
  __syncthreads();
  const float rinv = rinv_s;
  float* dstp = nodes + (size_t)dst * DIM;
  for (int d = threadIdx.x; d < DIM; d += blockDim.x) dstp[d] = src[d] * rinv;
}

// ---------------------------------------------------------------------------
// Kernel 2: one tree level of circular correlations via f32 WMMA.
// One 128-thread block (4 waves) per composition node.
//   D[m,n] = sum_j a_ext[j+m] * b_ext[j + base + 16n] = c[(base+16n-m) mod 1024]
// Wave w uses base = 15 + 256*w -> wave w covers c[256w .. 256w+255] exactly.
// ---------------------------------------------------------------------------
__global__ void corr_kernel(const int* __restrict__ comp, int off,
                            float* __restrict__ nodes) {
  __shared__ float a_ext[2 * DIM];
  __shared__ float b_ext[2 * DIM];
  __shared__ float cbuf[DIM];
  __shared__ float red[4];
  __shared__ float rinv_s;

  const int node = off + blockIdx.x;
  const int li = comp[3 * node + 0];
  const int ri = comp[3 * node + 1];
  const int pi = comp[3 * node + 2];
  const float* aG = nodes + (size_t)li * DIM;
  const float* bG = nodes + (size_t)ri * DIM;

  for (int d = threadIdx.x; d < DIM; d += blockDim.x) {
    float av = aG[d], bv = bG[d];
    a_ext[d] = av; a_ext[d + DIM] = av;
    b_ext[d] = bv; b_ext[d + DIM] = bv;
  }
  __syncthreads();

  const int lane = threadIdx.x & 31;
  const int wave = threadIdx.x >> 5;
  const int half = lane >> 4;   // selects K pair (ISA 16x4 f32 A layout)
  const int low  = lane & 15;   // A: row m    B: col n
  const int base = 15 + 256 * wave;

  // A operand: lane holds a_ext[t + 2*half + {0,1} + m]      (m = low)
  // B operand: lane holds b_ext[t + 2*half + {0,1} + base + 16n]  (n = low)
  const float* ap = &a_ext[2 * half + low];
  const float* bp = &b_ext[2 * half + base + 16 * low];

  v8f acc = {};
  #pragma unroll 8
  for (int t = 0; t < DIM; t += 4) {
    v2f A, B;
    A.x = ap[t]; A.y = ap[t + 1];
    B.x = bp[t]; B.y = bp[t + 1];
    acc = __builtin_amdgcn_wmma_f32_16x16x4_f32(
        /*neg_a=*/false, A, /*neg_b=*/false, B,
        /*c_mod=*/(short)0, acc, /*reuse_a=*/false, /*reuse_b=*/false);
  }

  // Scatter: C/D layout -> VGPR r, lanes0-15: M=r, lanes16-31: M=r+8; N=low.
  #pragma unroll
  for (int r = 0; r < 8; ++r) {
    int o = (base + 16 * low - (r + 8 * half)) & (DIM - 1);
    cbuf[o] = acc[r];
  }
  __syncthreads();

  // Normalize c and write parent.
  float ss = 0.0f;
  for (int d = threadIdx.x; d < DIM; d += blockDim.x) {
    float v = cbuf[d];
    ss += v * v;
  }
  ss = waveReduceSum(ss);
  if (lane == 0) red[wave] = ss;
  __syncthreads();
  if (threadIdx.x == 0) {
    float t = red[0] + red[1] + red[2] + red[3];
    rinv_s = 1.0f / sqrtf(t);
  }
  __syncthreads();
  const float rinv = rinv_s;
  float* dstp = nodes + (size_t)pi * DIM;
  for (int d = threadIdx.x; d < DIM; d += blockDim.x) dstp[d] = cbuf[d] * rinv;
}

// ---------------------------------------------------------------------------
// Kernel 3: out = sigmoid(nodes @ lin_w^T + lin_b), f32 WMMA GEMM.
// M = n_nodes (4095), N = C (512), K = 1024. One 16x16 tile per wave.
// ---------------------------------------------------------------------------
__global__ void gemm_kernel(const float* __restrict__ nodes,
                            const float* __restrict__ lin_w,
                            const float* __restrict__ lin_b,
                            float* __restrict__ out,
                            int n_nodes, int C) {
  const int lane = threadIdx.x & 31;
  const int wave = threadIdx.x >> 5;
  const int ntN = C >> 4;
  const int ntM = (n_nodes + 15) >> 4;
  const int tile = blockIdx.x * (blockDim.x >> 5) + wave;
  if (tile >= ntM * ntN) return;
  const int m0 = (tile / ntN) << 4;
  const int n0 = (tile % ntN) << 4;

  const int half = lane >> 4;
  const int low  = lane & 15;

  int arow = m0 + low;                       // A operand row (m = low)
  if (arow >= n_nodes) arow = n_nodes - 1;   // clamp; garbage rows not stored
  const int brow = n0 + low;                 // B operand col (n = low)

  // k-element offset inside row: 4t + 2*half (+0/+1) -> float2 index 2t (base +half)
  const float2* aP = (const float2*)(nodes + (size_t)arow * DIM) + half;
  const float2* bP = (const float2*)(lin_w + (size_t)brow * DIM) + half;

  v8f acc = {};
  #pragma unroll 8
  for (int t = 0; t < DIM / 4; ++t) {
    float2 av = aP[2 * t];
    float2 bv = bP[2 * t];
    v2f A; A.x = av.x; A.y = av.y;
    v2f B; B.x = bv.x; B.y = bv.y;
    acc = __builtin_amdgcn_wmma_f32_16x16x4_f32(
        false, A, false, B, (short)0, acc, false, false);
  }

  const int col = n0 + low;
  const float bias = lin_b[col];
  #pragma unroll
  for (int r = 0; r < 8; ++r) {
    int row = m0 + r + 8 * half;
    if (row < n_nodes) {
      float v = acc[r] + bias;
      out[(size_t)row * C + col] = 1.0f / (1.0f + expf(-v));
    }
  }
}

// ---------------------------------------------------------------------------
// Launcher: leaf gather/normalize, 11 dependent tree levels, final GEMM.
// All launches on `stream` (graph-capture safe; no mallocs/syncs).
// ---------------------------------------------------------------------------
extern "C" void kernel_launch(void* const* d_in, const int* in_sizes, int n_in,
                              void* d_out, int out_size, void* d_ws, size_t ws_size,
                              hipStream_t stream) {
  const int*   leaf_info = (const int*)d_in[0];   // (L,2) int32
  const int*   comp      = (const int*)d_in[1];   // (L-1,3) int32, bottom-up levels
  const float* emb_w     = (const float*)d_in[2]; // (V,1024)
  const float* lin_w     = (const float*)d_in[3]; // (C,1024)
  const float* lin_b     = (const float*)d_in[4]; // (C,)
  float*       out       = (float*)d_out;         // (2L-1, C)

  const int L = in_sizes[0] / 2;       // 2048
  const int C = in_sizes[4];           // 512
  const int n_nodes = 2 * L - 1;       // 4095

  float* nodes = (float*)d_ws;         // n_nodes * 1024 floats (~16.8 MB)
  if (ws_size < (size_t)n_nodes * DIM * sizeof(float)) return;

  leaf_kernel<<<L, 256, 0, stream>>>(leaf_info, emb_w, nodes);

  int off = 0;
  for (int cnt = L / 2; cnt >= 1; cnt >>= 1) {   // 11 dependent levels
    corr_kernel<<<cnt, 128, 0, stream>>>(comp, off, nodes);
    off += cnt;
  }

  const int ntM = (n_nodes + 15) / 16;           // 256
  const int ntN = C / 16;                        // 32
  const int tiles = ntM * ntN;                   // 8192 (one per wave)
  const int blocks = (tiles + 7) / 8;            // 8 waves / 256-thread block
  gemm_kernel<<<blocks, 256, 0, stream>>>(nodes, lin_w, lin_b, out, n_nodes, C);
}